// ObjectLoss_54382875902458
// MI455X (gfx1250) — compile-verified
//
#include <hip/hip_runtime.h>
#include <hip/hip_bf16.h>

typedef float v2f __attribute__((ext_vector_type(2)));
typedef float v8f __attribute__((ext_vector_type(8)));

#define BDIM    256
#define SPLITS  16          // blocks per sample
#define NB      32          // batch
#define NA      3           // anchors
#define NH      64
#define NW      64
#define NC      85
#define NT      64          // targets per sample
#define GRID    (NA * NH * NW)        // 12288
#define BMWORDS (GRID / 32)           // 384

__global__ void zero_out_kernel(float* o) { o[0] = 0.0f; }

__global__ __launch_bounds__(BDIM)
void ObjectLoss_54382875902458_kernel(const float* __restrict__ output,
                                      const float* __restrict__ targets,
                                      const float* __restrict__ anchors,
                                      float* __restrict__ out)
{
    __shared__ unsigned int bitmap[BMWORDS];
    __shared__ float wred[BDIM / 32];

    const int tid = threadIdx.x;
    const int b   = blockIdx.x / SPLITS;
    const int s   = blockIdx.x % SPLITS;

    // ---- zero the gt bitmap (12288 bits) ----
    for (int j = tid; j < BMWORDS; j += BDIM) bitmap[j] = 0u;
    __syncthreads();

    // ---- sparse gt construction: one lane per target ----
    if (tid < NT) {
        const float* tg = targets + ((size_t)b * NT + tid) * 5;
        const float bx = tg[1], by = tg[2], bw = tg[3], bh = tg[4];
        const bool valid = !(bx == 0.0f && by == 0.0f && bw == 0.0f && bh == 0.0f);
        const float txs = bx * (float)NW, tys = by * (float)NH;
        const float tws = bw * (float)NW, ths = bh * (float)NH;
        const float cx = floorf(txs), cy = floorf(tys);
        const float zx = txs - (cx + 0.5f), zy = tys - (cy + 0.5f);
        const float tx0 = zx - tws * 0.5f, tx1 = zx + tws * 0.5f;
        const float ty0 = zy - ths * 0.5f, ty1 = zy + ths * 0.5f;
        const float area_t = tws * ths;

        float best = -1.0f;
        int   bi   = 0;
        #pragma unroll
        for (int j = 0; j < NA; ++j) {
            const float aw = anchors[2 * j], ah = anchors[2 * j + 1];
            const float axh = aw * 0.5f, ayh = ah * 0.5f;
            const float x0 = fmaxf(tx0, -axh), x1 = fminf(tx1, axh);
            const float y0 = fmaxf(ty0, -ayh), y1 = fminf(ty1, ayh);
            const float inter = ((x0 < x1) && (y0 < y1)) ? (x1 - x0) * (y1 - y0) : 0.0f;
            const float iou = inter / (area_t + aw * ah - inter);
            if (iou > best) { best = iou; bi = j; }   // strict > == first-occurrence argmax
        }
        if (valid && best > 0.5f) {
            int cxi = (int)cx; cxi = cxi < 0 ? 0 : (cxi > NW - 1 ? NW - 1 : cxi);
            int cyi = (int)cy; cyi = cyi < 0 ? 0 : (cyi > NH - 1 ? NH - 1 : cyi);
            const int idx = bi * (NH * NW) + cyi * NW + cxi;   // (A,H,W) flat order
            atomicOr(&bitmap[idx >> 5], 1u << (idx & 31));
        }
    }
    __syncthreads();

    // ---- weighted BCE over this block's slice of the objectness channel ----
    const size_t base  = (size_t)b * ((size_t)NA * NH * NW * NC);
    const int    chunk = GRID / SPLITS;          // 768, divisible by BDIM -> uniform trip count
    const int    i0    = s * chunk;
    float acc = 0.0f;
    for (int i = i0 + tid; i < i0 + chunk; i += BDIM) {
        const float p  = __builtin_nontemporal_load(output + base + (size_t)i * NC + 4);
        const unsigned gt = (bitmap[i >> 5] >> (i & 31)) & 1u;
        const float lp  = fmaxf(logf(p),     -100.0f);
        const float l1p = fmaxf(log1pf(-p),  -100.0f);
        acc += gt ? -lp : -0.5f * l1p;           // weight*bce, gt in {0,1}
    }

    // ---- wave32 reduction on the matrix pipe: D = A(acc) x ones + 0 ----
    // A vgpr0 = acc, vgpr1 = 0  =>  D[m][n] = acc(m) + acc(m+16) for all n.
    // D layout: lane L<16 holds rows 0..7 in cv[0..7]; lanes>=16 hold rows 8..15.
    v2f av; av.x = acc;  av.y = 0.0f;
    v2f bv; bv.x = 1.0f; bv.y = 1.0f;
    v8f cv = {};
    cv = __builtin_amdgcn_wmma_f32_16x16x4_f32(
            /*neg_a=*/false, av, /*neg_b=*/false, bv,
            /*c_mod=*/(short)0, cv, /*reuse_a=*/false, /*reuse_b=*/false);
    float t = cv[0] + cv[1] + cv[2] + cv[3] + cv[4] + cv[5] + cv[6] + cv[7];
    const float wsum = t + __shfl_xor(t, 16, 32);     // rows 0..7 half + rows 8..15 half

    if ((tid & 31) == 0) wred[tid >> 5] = wsum;
    __syncthreads();
    if (tid == 0) {
        float ssum = 0.0f;
        #pragma unroll
        for (int w = 0; w < BDIM / 32; ++w) ssum += wred[w];
        // per-sample mean (1/GRID) and batch mean (1/NB) folded into one scale
        atomicAdd(out, ssum * (1.0f / ((float)GRID * (float)NB)));
    }
}

extern "C" void kernel_launch(void* const* d_in, const int* in_sizes, int n_in,
                              void* d_out, int out_size, void* d_ws, size_t ws_size,
                              hipStream_t stream) {
    const float* output  = (const float*)d_in[0];
    const float* targets = (const float*)d_in[1];
    const float* anchors = (const float*)d_in[2];
    float* out = (float*)d_out;

    zero_out_kernel<<<1, 1, 0, stream>>>(out);
    ObjectLoss_54382875902458_kernel<<<NB * SPLITS, BDIM, 0, stream>>>(
        output, targets, anchors, out);
}